// CoPEModel_19997367730788
// MI455X (gfx1250) — compile-verified
//
#include <hip/hip_runtime.h>

typedef __attribute__((ext_vector_type(2))) float v2f;
typedef __attribute__((ext_vector_type(8))) float v8f;

#define NF 128    // feature / hidden width
#define NC 16     // classifier width
#define BN_EPS 1e-5f
#define ASTR 132  // padded LDS row stride for A tiles (dwords)
#define WSTR 136  // padded LDS row stride for W (dwords)

// ---------------------------------------------------------------- degrees ---
__global__ void gcn_deg_kernel(const int* __restrict__ src, const int* __restrict__ dst,
                               float* __restrict__ outdeg, float* __restrict__ indeg, int E) {
    int e = blockIdx.x * blockDim.x + threadIdx.x;
    if (e < E) {
        atomicAdd(&outdeg[src[e]], 1.0f);
        atomicAdd(&indeg[dst[e]], 1.0f);
    }
}

__global__ void gcn_norm_kernel(float* __restrict__ outdeg, float* __restrict__ indeg, int N) {
    int i = blockIdx.x * blockDim.x + threadIdx.x;
    if (i < N) {
        outdeg[i] = rsqrtf(fmaxf(outdeg[i], 1.0f));   // clip(deg,1)^{-1/2}
        indeg[i]  = rsqrtf(fmaxf(indeg[i], 1.0f));
    }
}

// ------------------------------------------------------------ WMMA GEMM -----
// Y[m,:] = rownorm[m] * ( f(A[m,:]) @ W ),  f = relu(a*s+t) if APPLY_BN else a.
// rownorm >= 0 commutes with the K-reduction -> applied to accumulator rows.
// The 16x128 A tile is staged in LDS once per block with BN+ReLU pre-applied,
// so the unrolled K-loop is pure ds_load -> v_wmma_f32_16x16x4_f32 (32 steps).
// 8 waves/block cover all 128 output columns from the LDS-resident W.
template <bool APPLY_BN>
__global__ __launch_bounds__(256)
void gcn_gemm128_wmma(const float* __restrict__ A, const float* __restrict__ W,
                      const float* __restrict__ rownorm,
                      const float* __restrict__ bnscale, const float* __restrict__ bnshift,
                      float* __restrict__ Y, int N) {
    __shared__ float Ws[NF * WSTR];                    // ~68 KB, padded rows
    __shared__ float As[16 * ASTR];                    // ~8.3 KB, padded rows
    const int m0 = blockIdx.x * 16;

    // stage W (row-internal contiguity preserved; row base 544B -> 16B aligned)
    for (int i = threadIdx.x; i < (NF * NF) / 4; i += blockDim.x) {
        int r = i >> 5, c = (i & 31) << 2;
        *(float4*)&Ws[r * WSTR + c] = ((const float4*)W)[i];
    }
    // stage A tile, BN+ReLU applied once here (zero-fill OOB rows)
    for (int i = threadIdx.x; i < (16 * NF) / 4; i += blockDim.x) {
        int r = i >> 5, c = (i & 31) << 2;
        int m = m0 + r;
        float4 v;
        if (m < N) v = *(const float4*)(A + (size_t)m * NF + c);
        else       { v.x = 0.f; v.y = 0.f; v.z = 0.f; v.w = 0.f; }
        if (APPLY_BN) {
            float4 s = *(const float4*)(bnscale + c);
            float4 t = *(const float4*)(bnshift + c);
            v.x = fmaxf(v.x * s.x + t.x, 0.0f);
            v.y = fmaxf(v.y * s.y + t.y, 0.0f);
            v.z = fmaxf(v.z * s.z + t.z, 0.0f);
            v.w = fmaxf(v.w * s.w + t.w, 0.0f);
        }
        *(float4*)&As[r * ASTR + c] = v;
    }
    __syncthreads();

    const int wave = threadIdx.x >> 5;
    const int lane = threadIdx.x & 31;
    const int n0   = wave * 16;
    const int mr   = lane & 15;                        // row (A) / col (B,C) in tile
    const int kh   = (lane >> 4) << 1;                 // lanes 0-15 -> K 0,1 ; 16-31 -> K 2,3

    v8f acc = {};
    #pragma unroll
    for (int k = 0; k < NF; k += 4) {
        v2f a; a.x = As[mr * ASTR + k + kh];
               a.y = As[mr * ASTR + k + kh + 1];
        v2f b; b.x = Ws[(k + kh) * WSTR + n0 + mr];
               b.y = Ws[(k + kh + 1) * WSTR + n0 + mr];
        acc = __builtin_amdgcn_wmma_f32_16x16x4_f32(
                  false, a, false, b, (short)0, acc, false, false);
    }

    // C/D layout: VGPR i -> M = i (lanes 0-15) / i+8 (lanes 16-31), N = lane&15
    const int mbase = m0 + ((lane >> 4) << 3);
    const int col   = n0 + mr;
    #pragma unroll
    for (int i = 0; i < 8; ++i) {
        int m = mbase + i;
        if (m < N) Y[(size_t)m * NF + col] = acc[i] * rownorm[m];  // half-wave-uniform rn
    }
}

// ----------------------------------------------------- edge scatter (agg) ---
// agg[dst[e], :] += H[src[e], :]   ; float4 gather + 4 f32 atomics per thread
__global__ void gcn_scatter_kernel(const float* __restrict__ H, const int* __restrict__ src,
                                   const int* __restrict__ dst, float* __restrict__ agg,
                                   int E) {
    long long idx = (long long)blockIdx.x * blockDim.x + threadIdx.x;
    if (idx >= (long long)E * 32) return;
    int e = (int)(idx >> 5);
    int q = ((int)idx & 31) << 2;                      // channel group of 4
    int s = src[e], d = dst[e];
    float4 v = *(const float4*)(H + (size_t)s * NF + q);
    float* p = agg + (size_t)d * NF + q;
    atomicAdd(p + 0, v.x);
    atomicAdd(p + 1, v.y);
    atomicAdd(p + 2, v.z);
    atomicAdd(p + 3, v.w);
}

// -------------------------------------------- post-aggregate + BN stats ----
// h = agg*in_norm + b (in place), accumulate per-channel sum / sumsq.
// blockDim = 128 (one channel per thread), each block walks 256 rows,
// then one atomic per channel per block.
__global__ __launch_bounds__(128)
void gcn_post_bn_kernel(float* __restrict__ agg, const float* __restrict__ innorm,
                        const float* __restrict__ bias,
                        float* __restrict__ bn_sum, float* __restrict__ bn_ssq, int N) {
    const int c  = threadIdx.x;
    const int r0 = blockIdx.x * 256;
    const float b = bias[c];
    float ls = 0.0f, lss = 0.0f;
    for (int r = 0; r < 256; ++r) {
        int i = r0 + r;
        if (i >= N) break;
        float v = agg[(size_t)i * NF + c] * innorm[i] + b;
        agg[(size_t)i * NF + c] = v;
        ls += v; lss += v * v;
    }
    atomicAdd(&bn_sum[c], ls);
    atomicAdd(&bn_ssq[c], lss);
}

__global__ void gcn_bnfinal_kernel(const float* __restrict__ sum, const float* __restrict__ ssq,
                                   const float* __restrict__ gamma, const float* __restrict__ beta,
                                   float* __restrict__ s, float* __restrict__ t, int N) {
    int c = threadIdx.x;
    float invN = 1.0f / (float)N;
    float mu   = sum[c] * invN;
    float var  = ssq[c] * invN - mu * mu;
    float sc   = gamma[c] * rsqrtf(var + BN_EPS);
    s[c] = sc;
    t[c] = beta[c] - mu * sc;
}

// ------------------------------------------------------- classifier GEMM ---
// out[m, 0:16] = relu(H[m,:]*s + t) @ Wc + bc.
// 128-row A tile staged in LDS with BN+ReLU pre-applied; one wave per 16-row
// sub-tile; Wc (128x16) resident in LDS.
__global__ __launch_bounds__(256)
void gcn_classifier_wmma(const float* __restrict__ H, const float* __restrict__ Wc,
                         const float* __restrict__ bc,
                         const float* __restrict__ bnscale, const float* __restrict__ bnshift,
                         float* __restrict__ out, int N) {
    __shared__ float As[128 * ASTR];                   // ~67.6 KB
    __shared__ float Wcs[NF * NC];                     // 8 KB (no pad needed: +32-bank split)
    const int mblk = blockIdx.x * 128;

    for (int i = threadIdx.x; i < (NF * NC) / 4; i += blockDim.x)
        ((float4*)Wcs)[i] = ((const float4*)Wc)[i];
    for (int i = threadIdx.x; i < (128 * NF) / 4; i += blockDim.x) {
        int r = i >> 5, c = (i & 31) << 2;
        int m = mblk + r;
        float4 v;
        if (m < N) v = *(const float4*)(H + (size_t)m * NF + c);
        else       { v.x = 0.f; v.y = 0.f; v.z = 0.f; v.w = 0.f; }
        float4 s = *(const float4*)(bnscale + c);
        float4 t = *(const float4*)(bnshift + c);
        v.x = fmaxf(v.x * s.x + t.x, 0.0f);
        v.y = fmaxf(v.y * s.y + t.y, 0.0f);
        v.z = fmaxf(v.z * s.z + t.z, 0.0f);
        v.w = fmaxf(v.w * s.w + t.w, 0.0f);
        *(float4*)&As[r * ASTR + c] = v;
    }
    __syncthreads();

    const int wave = threadIdx.x >> 5;
    const int lane = threadIdx.x & 31;
    const int m0   = mblk + wave * 16;
    if (m0 >= N) return;                               // uniform per-wave exit

    const int mr = lane & 15;
    const int kh = (lane >> 4) << 1;
    const int ar = (wave * 16 + mr) * ASTR;

    v8f acc = {};
    #pragma unroll
    for (int k = 0; k < NF; k += 4) {
        v2f a; a.x = As[ar + k + kh];
               a.y = As[ar + k + kh + 1];
        v2f b; b.x = Wcs[(k + kh) * NC + mr];
               b.y = Wcs[(k + kh + 1) * NC + mr];
        acc = __builtin_amdgcn_wmma_f32_16x16x4_f32(
                  false, a, false, b, (short)0, acc, false, false);
    }

    const int mbase = m0 + ((lane >> 4) << 3);
    const float bias = bc[mr];
    #pragma unroll
    for (int i = 0; i < 8; ++i) {
        int m = mbase + i;
        if (m < N) out[(size_t)m * NC + mr] = acc[i] + bias;
    }
}

// -------------------------------------------------------------- launcher ---
extern "C" void kernel_launch(void* const* d_in, const int* in_sizes, int n_in,
                              void* d_out, int out_size, void* d_ws, size_t ws_size,
                              hipStream_t stream) {
    const float* x   = (const float*)d_in[0];
    const int*   src = (const int*)  d_in[1];
    const int*   dst = (const int*)  d_in[2];
    const float* W1  = (const float*)d_in[3];
    const float* b1  = (const float*)d_in[4];
    const float* g1  = (const float*)d_in[5];
    const float* be1 = (const float*)d_in[6];
    const float* W2  = (const float*)d_in[7];
    const float* b2  = (const float*)d_in[8];
    const float* g2  = (const float*)d_in[9];
    const float* be2 = (const float*)d_in[10];
    const float* Wc  = (const float*)d_in[11];
    const float* bc  = (const float*)d_in[12];

    const int N = in_sizes[0] / NF;
    const int E = in_sizes[1];

    float* w    = (float*)d_ws;
    float* outn = w;  w += N;                 // out-degree -> out-norm
    float* inn  = w;  w += N;                 // in-degree  -> in-norm
    float* bufA = w;  w += (size_t)N * NF;    // pre-scatter features (h1pre / h2pre)
    float* bufB = w;  w += (size_t)N * NF;    // aggregate -> gc output (h1 / h2)
    float* bnsum = w; w += NF;
    float* bnssq = w; w += NF;
    float* s1 = w; w += NF;
    float* t1 = w; w += NF;
    float* s2 = w; w += NF;
    float* t2 = w; w += NF;

    const int mtiles = (N + 15) / 16;
    const long long sthreads = (long long)E * 32;
    const unsigned sgrid = (unsigned)((sthreads + 255) / 256);

    // init: degrees, agg buffer, BN accumulators
    hipMemsetAsync(outn, 0, sizeof(float) * 2 * (size_t)N, stream);
    hipMemsetAsync(bufB, 0, sizeof(float) * (size_t)N * NF, stream);
    hipMemsetAsync(bnsum, 0, sizeof(float) * 2 * NF, stream);

    gcn_deg_kernel<<<(E + 255) / 256, 256, 0, stream>>>(src, dst, outn, inn, E);
    gcn_norm_kernel<<<(N + 255) / 256, 256, 0, stream>>>(outn, inn, N);

    // ---- layer 1 ----
    gcn_gemm128_wmma<false><<<mtiles, 256, 0, stream>>>(x, W1, outn, nullptr, nullptr, bufA, N);
    gcn_scatter_kernel<<<sgrid, 256, 0, stream>>>(bufA, src, dst, bufB, E);
    gcn_post_bn_kernel<<<(N + 255) / 256, 128, 0, stream>>>(bufB, inn, b1, bnsum, bnssq, N);
    gcn_bnfinal_kernel<<<1, NF, 0, stream>>>(bnsum, bnssq, g1, be1, s1, t1, N);

    // ---- layer 2 (BN1+ReLU applied during A-tile staging) ----
    gcn_gemm128_wmma<true><<<mtiles, 256, 0, stream>>>(bufB, W2, outn, s1, t1, bufA, N);
    hipMemsetAsync(bufB, 0, sizeof(float) * (size_t)N * NF, stream);   // recycle as agg2
    hipMemsetAsync(bnsum, 0, sizeof(float) * 2 * NF, stream);
    gcn_scatter_kernel<<<sgrid, 256, 0, stream>>>(bufA, src, dst, bufB, E);
    gcn_post_bn_kernel<<<(N + 255) / 256, 128, 0, stream>>>(bufB, inn, b2, bnsum, bnssq, N);
    gcn_bnfinal_kernel<<<1, NF, 0, stream>>>(bnsum, bnssq, g2, be2, s2, t2, N);

    // ---- classifier (BN2+ReLU applied during A-tile staging) ----
    gcn_classifier_wmma<<<(N + 127) / 128, 256, 0, stream>>>(bufB, Wc, bc, s2, t2,
                                                             (float*)d_out, N);
}